// InstanceRigidModel_3848290697507
// MI455X (gfx1250) — compile-verified
//
#include <hip/hip_runtime.h>
#include <math.h>

#define N_POSE     2048
#define K_PAIR     2096128
#define REG_W      0.001f
#define EPS_F      1e-6f
#define PI_F       3.14159265358979323846f

#define THREADS    256
#define POSE_BLOCKS 8      /* 8*256 == 2048, exact */
#define PAIR_BLOCKS 1024   /* 1024*256 = 262144 threads, ~8 pairs each */

typedef float v2f __attribute__((ext_vector_type(2)));
typedef float v8f __attribute__((ext_vector_type(8)));

// Full 32-lane sum using V_WMMA_F32_16X16X4_F32 (A=(p,0), B=ones):
// D[m][n] = p(lane m) + p(lane m+16); summing the 8 C/D vgprs gives rows 0-7
// (lanes 0-15) / rows 8-15 (lanes 16-31); one xor-16 shuffle finishes.
// Requires EXEC all-ones -> only call from fully-converged full waves.
__device__ __forceinline__ float wave_sum32(float p) {
  v2f a; a.x = p;    a.y = 0.0f;
  v2f b; b.x = 1.0f; b.y = 1.0f;
  v8f c = {};
  c = __builtin_amdgcn_wmma_f32_16x16x4_f32(false, a, false, b,
                                            (short)0, c, false, false);
  float s = ((c[0] + c[1]) + (c[2] + c[3])) + ((c[4] + c[5]) + (c[6] + c[7]));
  s += __shfl_xor(s, 16, 32);
  return s;
}

// acos for x in [-1+eps, 1-eps]: A&S 4.4.45, |err| ~3e-8 rad.
// sqrt(1-t)*poly(t) form is well-conditioned near |x|=1 (theta -> 0 / pi).
__device__ __forceinline__ float fast_acosf(float x) {
  float t = fabsf(x);
  float p = -0.0012624911f;
  p = fmaf(p, t,  0.0066700901f);
  p = fmaf(p, t, -0.0170881256f);
  p = fmaf(p, t,  0.0308918810f);
  p = fmaf(p, t, -0.0501743046f);
  p = fmaf(p, t,  0.0889789874f);
  p = fmaf(p, t, -0.2145988016f);
  p = fmaf(p, t,  1.5707963050f);
  float r = sqrtf(1.0f - t) * p;
  return (x < 0.0f) ? (PI_F - r) : r;
}

// SE(3) log matching the reference (reference uses ELEMENTWISE W*W in Vinv;
// W has zero diagonal, so Vinv diagonal == 1).
// Divisions replaced by v_rcp_f32; sin(th) folded via sin(th)=2*sin(hh)*cos(hh).
// Clamp bounds th away from {0, pi}: sin(hh), cos(hh) >= ~7e-4, rcp is safe.
__device__ __forceinline__ void se3_log(const float R[9],
                                        float t0, float t1, float t2,
                                        float l[6]) {
  float tr  = R[0] + R[4] + R[8];
  float ct  = fminf(fmaxf((tr - 1.0f) * 0.5f, -1.0f + EPS_F), 1.0f - EPS_F);
  float th  = fast_acosf(ct) + EPS_F;
  float hh  = 0.5f * th;
  float sh  = __sinf(hh);
  float chh = __cosf(hh);
  float inv_sh = __builtin_amdgcn_rcpf(sh);
  float inv_ch = __builtin_amdgcn_rcpf(chh);
  float inv_th = __builtin_amdgcn_rcpf(th);
  // scl = th / (2*sin(th)) = 0.25*th / (sin(hh)*cos(hh))
  float scl = 0.25f * th * inv_sh * inv_ch;
  // coef = (1 - th*cos(hh)/(2*sin(hh))) / th^2
  float coef = (1.0f - 0.5f * th * chh * inv_sh) * inv_th * inv_th;

  float W01 = scl * (R[1] - R[3]);   // W[0][1]
  float W02 = scl * (R[2] - R[6]);   // W[0][2]
  float W12 = scl * (R[5] - R[7]);   // W[1][2]
  float q01 = coef * W01 * W01;
  float q02 = coef * W02 * W02;
  float q12 = coef * W12 * W12;
  // w = [W21, W02, W10]
  l[0] = -W12;
  l[1] =  W02;
  l[2] = -W01;
  // v = Vinv @ t,  Vinv = I - 0.5*W + coef*(W.W)
  l[3] = t0 + (-0.5f * W01 + q01) * t1 + (-0.5f * W02 + q02) * t2;
  l[4] = (0.5f * W01 + q01) * t0 + t1 + (-0.5f * W12 + q12) * t2;
  l[5] = (0.5f * W02 + q02) * t0 + (0.5f * W12 + q12) * t1 + t2;
}

__device__ __forceinline__ void block_emit(float wsum, float* out) {
  __shared__ float sm[THREADS / 32];
  int lane = threadIdx.x & 31;
  int wid  = threadIdx.x >> 5;
  if (lane == 0) sm[wid] = wsum;
  __syncthreads();
  if (threadIdx.x == 0) {
    float s = 0.0f;
#pragma unroll
    for (int w = 0; w < THREADS / 32; ++w) s += sm[w];
    out[blockIdx.x] = s;
  }
}

// Kernel 1: per-pose transforms (Ti, analytic SE3 inverse) + regularizer log.
__global__ void __launch_bounds__(THREADS)
pose_kernel(const float* __restrict__ angle, const float* __restrict__ trans,
            float* __restrict__ Ti, float* __restrict__ Tinv,
            float* __restrict__ regBlock) {
  const int n = blockIdx.x * THREADS + threadIdx.x;  // always < N_POSE
  const float DEG = PI_F / 180.0f;
  float a0 = angle[n]            * DEG;
  float a1 = angle[N_POSE + n]   * DEG;
  float a2 = angle[2*N_POSE + n] * DEG;
  float c0 = __cosf(a0), s0 = __sinf(a0);
  float c1 = __cosf(a1), s1 = __sinf(a1);
  float c2 = __cosf(a2), s2 = __sinf(a2);

  float R[9];
  R[0] = c2*c1;  R[1] = c2*s1*s0 - s2*c0;  R[2] = c2*s1*c0 + s2*s0;
  R[3] = s2*c1;  R[4] = s2*s1*s0 + c2*c0;  R[5] = s2*s1*c0 - c2*s0;
  R[6] = -s1;    R[7] = c1*s0;             R[8] = c1*c0;

  float t0 = trans[n], t1 = trans[N_POSE + n], t2 = trans[2*N_POSE + n];

  float4* Tp = (float4*)(Ti + n * 12);
  Tp[0] = make_float4(R[0], R[1], R[2], t0);
  Tp[1] = make_float4(R[3], R[4], R[5], t1);
  Tp[2] = make_float4(R[6], R[7], R[8], t2);

  // inv(Ti) = [R^T | -R^T t]
  float b0 = -(R[0]*t0 + R[3]*t1 + R[6]*t2);
  float b1 = -(R[1]*t0 + R[4]*t1 + R[7]*t2);
  float b2 = -(R[2]*t0 + R[5]*t1 + R[8]*t2);
  float4* Ip = (float4*)(Tinv + n * 12);
  Ip[0] = make_float4(R[0], R[3], R[6], b0);
  Ip[1] = make_float4(R[1], R[4], R[7], b1);
  Ip[2] = make_float4(R[2], R[5], R[8], b2);

  float l[6];
  se3_log(R, t0, t1, t2, l);
  float reg = l[0]*l[0] + l[1]*l[1] + l[2]*l[2]
            + l[3]*l[3] + l[4]*l[4] + l[5]*l[5];

  block_emit(wave_sum32(reg), regBlock);
}

// Kernel 2: per-pair loss, grid-stride, WMMA wave reduce, per-block sums.
__global__ void __launch_bounds__(THREADS)
pair_kernel(const float* __restrict__ logRobs,
            const int* __restrict__ pi, const int* __restrict__ pj,
            const float* __restrict__ Ti, const float* __restrict__ Tinv,
            float* __restrict__ blockSum) {
  const int tid    = blockIdx.x * THREADS + threadIdx.x;
  const int stride = PAIR_BLOCKS * THREADS;
  float acc = 0.0f;

#pragma unroll 2
  for (int k = tid; k < K_PAIR; k += stride) {
    int i = __builtin_nontemporal_load(&pi[k]);
    int j = __builtin_nontemporal_load(&pj[k]);
    const float4* Aq = (const float4*)(Ti   + j * 12);  // Tj rows (L2-resident)
    const float4* Bq = (const float4*)(Tinv + i * 12);  // inv(Ti) rows
    float4 A0 = Aq[0], A1 = Aq[1], A2 = Aq[2];
    float4 B0 = Bq[0], B1 = Bq[1], B2 = Bq[2];

    // Tij = Tj @ inv(Ti), rows 0..2 (row 3 trivial)
    float R[9], u0, u1, u2;
    R[0] = A0.x*B0.x + A0.y*B1.x + A0.z*B2.x;
    R[1] = A0.x*B0.y + A0.y*B1.y + A0.z*B2.y;
    R[2] = A0.x*B0.z + A0.y*B1.z + A0.z*B2.z;
    u0   = A0.x*B0.w + A0.y*B1.w + A0.z*B2.w + A0.w;
    R[3] = A1.x*B0.x + A1.y*B1.x + A1.z*B2.x;
    R[4] = A1.x*B0.y + A1.y*B1.y + A1.z*B2.y;
    R[5] = A1.x*B0.z + A1.y*B1.z + A1.z*B2.z;
    u1   = A1.x*B0.w + A1.y*B1.w + A1.z*B2.w + A1.w;
    R[6] = A2.x*B0.x + A2.y*B1.x + A2.z*B2.x;
    R[7] = A2.x*B0.y + A2.y*B1.y + A2.z*B2.y;
    R[8] = A2.x*B0.z + A2.y*B1.z + A2.z*B2.z;
    u2   = A2.x*B0.w + A2.y*B1.w + A2.z*B2.w + A2.w;

    float l[6];
    se3_log(R, u0, u1, u2, l);

    float d0 = l[0] - __builtin_nontemporal_load(&logRobs[k]);
    float d1 = l[1] - __builtin_nontemporal_load(&logRobs[K_PAIR   + k]);
    float d2 = l[2] - __builtin_nontemporal_load(&logRobs[2*K_PAIR + k]);
    float d3 = l[3] - __builtin_nontemporal_load(&logRobs[3*K_PAIR + k]);
    float d4 = l[4] - __builtin_nontemporal_load(&logRobs[4*K_PAIR + k]);
    float d5 = l[5] - __builtin_nontemporal_load(&logRobs[5*K_PAIR + k]);
    float ss = d0*d0 + d1*d1 + d2*d2 + d3*d3 + d4*d4 + d5*d5;
    acc += sqrtf(ss);
  }

  // Tail divergence reconverges before here; all waves are full -> EXEC all-1s.
  block_emit(wave_sum32(acc), blockSum);
}

// Kernel 3: deterministic final reduction (fixed-order tree, no atomics).
__global__ void __launch_bounds__(THREADS)
finalize_kernel(const float* __restrict__ blockSum,
                const float* __restrict__ regBlock,
                float* __restrict__ out) {
  __shared__ float sm[THREADS];
  float s = 0.0f;
  for (int b = threadIdx.x; b < PAIR_BLOCKS; b += THREADS) s += blockSum[b];
  sm[threadIdx.x] = s;
  __syncthreads();
  for (int off = THREADS / 2; off > 0; off >>= 1) {
    if (threadIdx.x < off) sm[threadIdx.x] += sm[threadIdx.x + off];
    __syncthreads();
  }
  if (threadIdx.x == 0) {
    float reg = 0.0f;
#pragma unroll
    for (int b = 0; b < POSE_BLOCKS; ++b) reg += regBlock[b];
    const float invK = 1.0f / (float)K_PAIR;
    out[0] = sm[0] * invK + REG_W * reg * invK;
  }
}

extern "C" void kernel_launch(void* const* d_in, const int* in_sizes, int n_in,
                              void* d_out, int out_size, void* d_ws, size_t ws_size,
                              hipStream_t stream) {
  const float* logRobs     = (const float*)d_in[0];
  const float* angle       = (const float*)d_in[1];
  const float* translation = (const float*)d_in[2];
  const int*   pair_i      = (const int*)d_in[3];
  const int*   pair_j      = (const int*)d_in[4];

  float* ws        = (float*)d_ws;
  float* pairBlock = ws;                       // [PAIR_BLOCKS]
  float* regBlock  = ws + PAIR_BLOCKS;         // [POSE_BLOCKS]
  float* Ti        = ws + 1040;                // [N_POSE*12], 16B aligned
  float* Tinv      = Ti + N_POSE * 12;         // [N_POSE*12]

  pose_kernel<<<POSE_BLOCKS, THREADS, 0, stream>>>(angle, translation,
                                                   Ti, Tinv, regBlock);
  pair_kernel<<<PAIR_BLOCKS, THREADS, 0, stream>>>(logRobs, pair_i, pair_j,
                                                   Ti, Tinv, pairBlock);
  finalize_kernel<<<1, THREADS, 0, stream>>>(pairBlock, regBlock, (float*)d_out);
}